// AggregateClusteredSum_2456721293667
// MI455X (gfx1250) — compile-verified
//
#include <hip/hip_runtime.h>

typedef __attribute__((ext_vector_type(2))) float v2f;
typedef __attribute__((ext_vector_type(8))) float v8f;
typedef __attribute__((ext_vector_type(4))) int   v4i;

#define AS1 __attribute__((address_space(1)))
#define AS3 __attribute__((address_space(3)))

#define KSEG 64
#define HD   128
#define HIDD 256
#define BQ   32
#define NQ   8192
#define RPB  129              // rows per batch in Hcat (2K+1)
#define MTOT (BQ * RPB)       // 4128 = 129 blocks * 32 rows (exact tiling)

// workspace layout (float offsets)
#define WS_SUMS 0                              // BQ*KSEG*HD floats
#define WS_CNT  (WS_SUMS + BQ * KSEG * HD)     // BQ*KSEG ints
#define WS_KS   (WS_CNT + BQ * KSEG)           // BQ ints
#define WS_ACTA (WS_KS + 64)                   // MTOT*HIDD floats
#define WS_ACTB (WS_ACTA + MTOT * HIDD)        // MTOT*HIDD floats

__global__ void k_init(float* __restrict__ sums, int* __restrict__ cnt,
                       int* __restrict__ ks) {
  int i = blockIdx.x * blockDim.x + threadIdx.x;
  if (i < BQ * KSEG * HD) sums[i] = 0.0f;
  if (i < BQ * KSEG) cnt[i] = 0;
  if (i < BQ) ks[i] = -1;
}

// Segment-sum: grid (BQ, NQ/256). LDS-privatized accumulation, global atomic flush.
__global__ __launch_bounds__(256)
void k_segsum(const float* __restrict__ hs, const int* __restrict__ cs,
              const int* __restrict__ n_ptr,
              float* __restrict__ gsum, int* __restrict__ gcnt,
              int* __restrict__ gks) {
  __shared__ float ssum[KSEG * HD];   // 32 KB
  __shared__ int scnt[KSEG];
  __shared__ int sks;
  const int tid = threadIdx.x, lane = tid & 31, wid = tid >> 5;
  const int b = blockIdx.x;
  const int base = blockIdx.y * 256;
  const int n = n_ptr[0];

  for (int i = tid; i < KSEG * HD; i += 256) ssum[i] = 0.0f;
  if (tid < KSEG) scnt[tid] = 0;
  if (tid == 0) sks = (int)0x80000000;
  __syncthreads();

  const float4* hs4 = (const float4*)hs;
  for (int r = 0; r < 32; ++r) {
    int j = base + wid * 32 + r;
    if (j >= n) break;                 // j ascending within the wave
    int c = cs[(size_t)b * NQ + j];
    if (lane == 0) {
      atomicMax(&sks, c);
      if ((unsigned)c < KSEG) atomicAdd(&scnt[c], 1);
    }
    if ((unsigned)c < KSEG) {
      float4 v = hs4[((size_t)b * NQ + j) * 32 + lane];
      int o = c * HD + lane * 4;
      atomicAdd(&ssum[o + 0], v.x);    // ds_add_f32
      atomicAdd(&ssum[o + 1], v.y);
      atomicAdd(&ssum[o + 2], v.z);
      atomicAdd(&ssum[o + 3], v.w);
    }
  }
  __syncthreads();
  for (int i = tid; i < KSEG * HD; i += 256) {
    float v = ssum[i];
    if (v != 0.0f) atomicAdd(&gsum[(size_t)b * KSEG * HD + i], v);
  }
  if (tid < KSEG && scnt[tid] != 0) atomicAdd(&gcnt[b * KSEG + tid], scnt[tid]);
  if (tid == 0) atomicMax(&gks[b], sks);
}

// Build Hcat (MTOT x 128) into actA with row stride HIDD.
__global__ __launch_bounds__(256)
void k_hcat(const float* __restrict__ sums, const float* __restrict__ hs,
            const int* __restrict__ n_ptr, float* __restrict__ actA) {
  const int tid = threadIdx.x, lane = tid & 31, wid = tid >> 5;
  const int row = blockIdx.x * 8 + wid;
  if (row >= MTOT) return;
  const int n = n_ptr[0];
  const int bq = row / RPB, rr = row - bq * RPB;
  const float4* hs4 = (const float4*)hs;
  const float4* s4 = (const float4*)sums;
  float4 hn = hs4[((size_t)bq * NQ + n) * 32 + lane];
  float4 out;
  if (rr < KSEG) {
    out = s4[((size_t)bq * KSEG + rr) * 32 + lane];
  } else if (rr < 2 * KSEG) {
    float4 hk = s4[((size_t)bq * KSEG + (rr - KSEG)) * 32 + lane];
    out.x = hk.x + hn.x; out.y = hk.y + hn.y;
    out.z = hk.z + hn.z; out.w = hk.w + hn.w;
  } else {
    out = hn;
  }
  ((float4*)actA)[(size_t)row * (HIDD / 4) + lane] = out;
}

// Fused GEMM layer: Out[MTOT x Nd] = act(A[MTOT x Kd] @ W[Kd x Nd] + b), fp32 WMMA 16x16x4.
template <int Kd, int Nd, bool PRELU, bool MASK>
__global__ __launch_bounds__(256)
void k_gemm(const float* __restrict__ A, int lda,
            const float* __restrict__ W, const float* __restrict__ bias,
            float* __restrict__ Out, int ldo,
            const float* __restrict__ alpha_ptr,
            const int* __restrict__ cnt) {
  // Pad by 4: rows stay 16B-aligned (STR*4 % 16 == 0) and column reads across
  // 16 consecutive rows hit 16 distinct banks (STR % 64 == 4).
  constexpr int STR = Kd + 4;
  constexpr int TN = Nd / 64;          // 16-wide col tiles per wave (wave strip = Nd/4)
  __shared__ float As[32 * STR];       // <= 32 * 260 * 4B = 33.3 KB

  const int tid = threadIdx.x, lane = tid & 31, wid = tid >> 5;
  const int m0 = blockIdx.x * 32;

  // Stage A tile (32 x Kd) via CDNA5 async global->LDS B128 copies (ASYNCcnt).
  {
    constexpr int NCHUNK = 32 * Kd / 4;          // float4 chunks
    constexpr int CPR = Kd / 4;                  // chunks per row
#pragma unroll
    for (int q = tid; q < NCHUNK; q += 256) {
      int r = q / CPR, c4 = q - r * CPR;
      const float* gsrc = A + (size_t)(m0 + r) * lda + c4 * 4;
      float* ldst = &As[r * STR + c4 * 4];
#if __has_builtin(__builtin_amdgcn_global_load_async_to_lds_b128)
      __builtin_amdgcn_global_load_async_to_lds_b128(
          (AS1 v4i*)gsrc, (AS3 v4i*)ldst, 0, 0);
#else
      *(float4*)ldst = *(const float4*)gsrc;
#endif
    }
#if __has_builtin(__builtin_amdgcn_global_load_async_to_lds_b128)
#if __has_builtin(__builtin_amdgcn_s_wait_asynccnt)
    __builtin_amdgcn_s_wait_asynccnt(0);
#else
    asm volatile("s_wait_asynccnt 0x0" ::: "memory");
#endif
#endif
  }
  __syncthreads();

  const int mw = wid & 1;              // 2 row-waves x 4 col-waves
  const int nw = wid >> 1;
  const int nbase = nw * (Nd / 4);
  const int l15 = lane & 15;
  const int hi = lane >> 4;            // 0: holds k,k+1 ; 1: holds k+2,k+3
  const int arow = mw * 16 + l15;

  v8f acc[TN];
  v8f zero = {0.f, 0.f, 0.f, 0.f, 0.f, 0.f, 0.f, 0.f};
#pragma unroll
  for (int t = 0; t < TN; ++t) acc[t] = zero;

  for (int kk = 0; kk < Kd; kk += 4) {
    // A fragment (16x4): lane<16 -> k,k+1 ; lane>=16 -> k+2,k+3 of row (mw*16+l15)
    v2f a;
    a.x = As[arow * STR + kk + 2 * hi];
    a.y = As[arow * STR + kk + 2 * hi + 1];
#pragma unroll
    for (int t = 0; t < TN; ++t) {
      int col = nbase + t * 16 + l15;
      // B fragment (4x16): lane<16 -> rows k,k+1 ; lane>=16 -> rows k+2,k+3
      v2f bf;
      bf.x = W[(size_t)(kk + 2 * hi) * Nd + col];
      bf.y = W[(size_t)(kk + 2 * hi + 1) * Nd + col];
      acc[t] = __builtin_amdgcn_wmma_f32_16x16x4_f32(
          false, a, false, bf, (short)0, acc[t], false, false);
    }
  }

  float alpha = 0.0f;
  if (PRELU) alpha = alpha_ptr[0];
#pragma unroll
  for (int t = 0; t < TN; ++t) {
    int col = nbase + t * 16 + l15;
    float bv = bias[col];
#pragma unroll
    for (int i = 0; i < 8; ++i) {
      int row = m0 + mw * 16 + i + 8 * hi;   // C/D: vgpr i -> M=i (lanes 0-15), M=i+8 (16-31)
      float v = acc[t][i] + bv;
      if (PRELU) v = (v >= 0.0f) ? v : alpha * v;
      if (MASK) {
        int bq = row / RPB, rr = row - bq * RPB;
        float mval = 1.0f;
        if (rr < KSEG)          mval = (cnt[bq * KSEG + rr] > 0) ? 1.0f : 0.0f;
        else if (rr < 2 * KSEG) mval = (cnt[bq * KSEG + rr - KSEG] > 0) ? 1.0f : 0.0f;
        v *= mval;
      }
      Out[(size_t)row * ldo + col] = v;
    }
  }
}

// Finish: S, Gk/GK combine, Ks move/zero logic, G_mask.
__global__ __launch_bounds__(128)
void k_final(const float* __restrict__ gs, const int* __restrict__ cnt,
             const int* __restrict__ ksarr, float* __restrict__ G,
             float* __restrict__ Gmask) {
  const int b = blockIdx.x, d = threadIdx.x;   // 128 threads: one per feature dim
  float S = 0.0f;
  for (int r = 0; r < KSEG; ++r) S += gs[((size_t)b * RPB + r) * HD + d];
  const float GK = S + gs[((size_t)b * RPB + 2 * KSEG) * HD + d];
  const int Ks = ksarr[b];
  const bool small = (Ks <= KSEG - 2);
  for (int j = 0; j <= KSEG; ++j) {
    float val;
    if (j < KSEG) {
      float ex = (cnt[b * KSEG + j] > 0) ? 1.0f : 0.0f;
      val = (S - gs[((size_t)b * RPB + j) * HD + d]
               + gs[((size_t)b * RPB + KSEG + j) * HD + d]) * ex;
    } else {
      val = GK;
    }
    if (small && j == Ks + 1) val = GK;
    if (small && j == KSEG) val = 0.0f;
    G[((size_t)b * (KSEG + 1) + j) * HD + d] = val;
  }
  if (d <= KSEG) {
    Gmask[b * (KSEG + 1) + d] = (small && d >= Ks + 2) ? 0.0f : 1.0f;
  }
}

extern "C" void kernel_launch(void* const* d_in, const int* in_sizes, int n_in,
                              void* d_out, int out_size, void* d_ws, size_t ws_size,
                              hipStream_t stream) {
  const float* hs = (const float*)d_in[0];
  const int* cs = (const int*)d_in[1];
  const int* n_ptr = (const int*)d_in[2];
  const float* W1 = (const float*)d_in[3];  const float* b1 = (const float*)d_in[4];
  const float* W2 = (const float*)d_in[5];  const float* b2 = (const float*)d_in[6];
  const float* W3 = (const float*)d_in[7];  const float* b3 = (const float*)d_in[8];
  const float* W4 = (const float*)d_in[9];  const float* b4 = (const float*)d_in[10];
  const float* W5 = (const float*)d_in[11]; const float* b5 = (const float*)d_in[12];
  const float* W6 = (const float*)d_in[13]; const float* b6 = (const float*)d_in[14];
  const float* a1 = (const float*)d_in[15];
  const float* a2 = (const float*)d_in[16];
  const float* a3 = (const float*)d_in[17];
  const float* a4 = (const float*)d_in[18];
  const float* a5 = (const float*)d_in[19];

  float* ws = (float*)d_ws;
  float* sums = ws + WS_SUMS;
  int* cnt = (int*)(ws + WS_CNT);
  int* ks = (int*)(ws + WS_KS);
  float* actA = ws + WS_ACTA;
  float* actB = ws + WS_ACTB;

  float* G = (float*)d_out;
  float* Gmask = G + (size_t)BQ * (KSEG + 1) * HD;

  k_init<<<(BQ * KSEG * HD + 255) / 256, 256, 0, stream>>>(sums, cnt, ks);
  k_segsum<<<dim3(BQ, NQ / 256), 256, 0, stream>>>(hs, cs, n_ptr, sums, cnt, ks);
  k_hcat<<<(MTOT + 7) / 8, 256, 0, stream>>>(sums, hs, n_ptr, actA);

  const int gx = MTOT / 32;   // 129 blocks, exact
  k_gemm<128, 256, true,  false><<<gx, 256, 0, stream>>>(actA, HIDD, W1, b1, actB, HIDD, a1, cnt);
  k_gemm<256, 256, true,  false><<<gx, 256, 0, stream>>>(actB, HIDD, W2, b2, actA, HIDD, a2, cnt);
  k_gemm<256, 256, true,  false><<<gx, 256, 0, stream>>>(actA, HIDD, W3, b3, actB, HIDD, a3, cnt);
  k_gemm<256, 256, true,  false><<<gx, 256, 0, stream>>>(actB, HIDD, W4, b4, actA, HIDD, a4, cnt);
  k_gemm<256, 256, true,  false><<<gx, 256, 0, stream>>>(actA, HIDD, W5, b5, actB, HIDD, a5, cnt);
  k_gemm<256, 128, false, true ><<<gx, 256, 0, stream>>>(actB, HIDD, W6, b6, actA, HD, nullptr, cnt);

  k_final<<<BQ, 128, 0, stream>>>(actA, cnt, ks, G, Gmask);
}